// LinearActor_2035814498408
// MI455X (gfx1250) — compile-verified
//
#include <hip/hip_runtime.h>
#include <math.h>

// Problem constants from the reference
constexpr int B = 64;
constexpr int S = 2048;
constexpr int D = 256;

typedef __attribute__((ext_vector_type(2))) float v2f;
typedef __attribute__((ext_vector_type(8))) float v8f;

// ---------------------------------------------------------------------------
// Phase 1: score[b,s] = dot(inputs[b,s,:], w) + bias, via V_WMMA_F32_16X16X4_F32.
// One wave = one 16-row tile. A fragment (16x4 f32, ISA layout):
//   lanes 0-15  : M = lane,    VGPR0=K0, VGPR1=K1
//   lanes 16-31 : M = lane-16, VGPR0=K2, VGPR1=K3
// B fragment (4x16 f32): w replicated across all N columns -> every output
// column of the 16x16 accumulator holds the same exact-f32 row score.
// ---------------------------------------------------------------------------
__global__ __launch_bounds__(256) void score_gemv_wmma(
    const float* __restrict__ x, const float* __restrict__ w,
    const float* __restrict__ bias, float* __restrict__ score)
{
    const int lane = threadIdx.x & 31;
    const int wave = threadIdx.x >> 5;
    const long tile = (long)blockIdx.x * 8 + wave;   // 16 rows per tile
    const long rowBase = tile * 16;
    const int  mrow = lane & 15;
    const int  koff = (lane >> 4) << 1;              // 0 for lanes 0-15, 2 for 16-31

    const float* rowPtr = x + (rowBase + mrow) * D + koff;
    const float* wPtr   = w + koff;

    v8f c = {};
#pragma unroll 8
    for (int d0 = 0; d0 < D; d0 += 4) {
        v2f a  = *(const v2f*)(rowPtr + d0);   // K = koff, koff+1
        v2f bf = *(const v2f*)(wPtr + d0);     // same K slice of w, all N columns equal
        c = __builtin_amdgcn_wmma_f32_16x16x4_f32(
                /*neg_a=*/false, a, /*neg_b=*/false, bf,
                /*c_mod=*/(short)0, c, /*reuse_a=*/false, /*reuse_b=*/false);
    }

    const float b0 = bias[0];
    const int   hi = (lane >> 4) << 3;               // lanes 16-31 hold M = r+8
#pragma unroll
    for (int r = 0; r < 8; ++r) {
        if ((lane & 15) == r)
            score[rowBase + r + hi] = c[r] + b0;     // any N column: all identical
    }
}

// ---------------------------------------------------------------------------
// Phase 2: per batch row, suffix logaddexp over score[b, guide[b,t]].
// lp[b,t] = v[t] - logaddexp(v[t..S-1]).  Reversed -> inclusive prefix scan.
// One 256-thread block per row, 8 elements/thread, Hillis-Steele in LDS.
// ---------------------------------------------------------------------------
__device__ __forceinline__ float lae(float a, float b) {
    float m = fmaxf(a, b);
    float d = fabsf(a - b);              // inf when one side is -inf -> exp->0
    return m + log1pf(expf(-d));
}

__global__ __launch_bounds__(256) void suffix_logaddexp_scan(
    const float* __restrict__ score, const int* __restrict__ guide,
    float* __restrict__ lp)
{
    __shared__ float sdata[256];
    constexpr int E = S / 256;           // 8
    const int b = blockIdx.x;
    const int i = threadIdx.x;

    const float* srow = score + (long)b * S;
    const int*   grow = guide + (long)b * S;

    float v[E], p[E];
    float run = -__builtin_inff();
#pragma unroll
    for (int j = 0; j < E; ++j) {
        int idx = i * E + j;             // reversed position
        int t   = S - 1 - idx;
        v[j] = srow[grow[t]];
        run  = lae(run, v[j]);
        p[j] = run;                      // local inclusive prefix (reversed)
    }

    float x = run;
    sdata[i] = x;
    __syncthreads();
    for (int off = 1; off < 256; off <<= 1) {
        float y = (i >= off) ? sdata[i - off] : -__builtin_inff();
        __syncthreads();
        x = lae(x, y);
        sdata[i] = x;
        __syncthreads();
    }
    float excl = (i > 0) ? sdata[i - 1] : -__builtin_inff();

    float* orow = lp + (long)b * S;
#pragma unroll
    for (int j = 0; j < E; ++j) {
        int   idx = i * E + j;
        float P   = lae(excl, p[j]);     // suffix-LSE at step t = S-1-idx
        orow[S - 1 - idx] = v[j] - P;
    }
}

// Second tuple output: guide, bit-copied after the B*S logprobs.
__global__ __launch_bounds__(256) void copy_guide(
    const int* __restrict__ g, int* __restrict__ out, int n)
{
    int k = blockIdx.x * blockDim.x + threadIdx.x;
    if (k < n) out[k] = g[k];
}

extern "C" void kernel_launch(void* const* d_in, const int* in_sizes, int n_in,
                              void* d_out, int out_size, void* d_ws, size_t ws_size,
                              hipStream_t stream) {
    const float* x     = (const float*)d_in[0];   // [B,S,D] f32
    const float* w     = (const float*)d_in[1];   // [D] f32
    const float* bias  = (const float*)d_in[2];   // [1] f32
    const int*   guide = (const int*)  d_in[3];   // [B,S] i32

    float* out   = (float*)d_out;                 // [B,S] lp  ++  [B,S] guide bits
    float* score = (float*)d_ws;                  // B*S floats = 512 KB scratch

    // Phase 1: 8192 tiles of 16 rows, 8 waves (256 thr) per block -> 1024 blocks
    score_gemv_wmma<<<(B * S / 16) / 8, 256, 0, stream>>>(x, w, bias, score);

    // Phase 2: one block per batch row
    suffix_logaddexp_scan<<<B, 256, 0, stream>>>(score, guide, out);

    // Tuple output #2
    copy_guide<<<(B * S + 255) / 256, 256, 0, stream>>>(
        guide, (int*)(out + (long)B * S), B * S);
}